// EnergyAttentionImage_47974784697012
// MI455X (gfx1250) — compile-verified
//
#include <hip/hip_runtime.h>
#include <math.h>

#define HH 512
#define WW 512
#define HW (HH*WW)
#define NB 16

typedef float v2f __attribute__((ext_vector_type(2)));
typedef float v8f __attribute__((ext_vector_type(8)));

__device__ __forceinline__ float softplusf(float x){
  return (x > 20.0f) ? x : log1pf(expf(x));
}

// ---------------------------------------------------------------------------
// WMMA band-matrix vertical 9-tap box sum.
// For one 16x16 output subtile: D[m][n] = sum_{k=m..m+8} hs[k][n]
// computed as D = A(16x24, 0/1 band) x B(24x16) via 6 chained
// V_WMMA_F32_16X16X4_F32 ops (full f32 precision, matrix pipe).
// A layout (ISA 7.12.2, 32-bit A 16x4): lanes 0-15 rows M=0..15 {K0,K1},
// lanes 16-31 same rows {K2,K3}. B layout: rows striped across VGPRs,
// lanes = columns. D: VGPR r -> row r (lanes 0-15) / row r+8 (lanes 16-31).
// ---------------------------------------------------------------------------
__device__ __forceinline__ void box9_vert_wmma(const float* hs, int hsPitch,
                                               float* out, int outPitch,
                                               int lane)
{
  const int n    = lane & 15;   // column for B/D, row for A
  const int half = lane >> 4;
  v8f acc = {0.f,0.f,0.f,0.f,0.f,0.f,0.f,0.f};
#pragma unroll
  for (int kc = 0; kc < 6; ++kc){
    const int kb = kc*4 + half*2;
    v2f a, bm;
    a.x = (kb   >= n && kb   <= n+8) ? 1.0f : 0.0f;  // band: k in [m, m+8]
    a.y = (kb+1 >= n && kb+1 <= n+8) ? 1.0f : 0.0f;
    bm.x = hs[(kb  )*hsPitch + n];
    bm.y = hs[(kb+1)*hsPitch + n];
    acc = __builtin_amdgcn_wmma_f32_16x16x4_f32(false, a, false, bm,
                                                (short)0, acc, false, false);
  }
#pragma unroll
  for (int r = 0; r < 8; ++r)
    out[(r + half*8)*outPitch + n] = acc[r];
}

// ---------------------------------------------------------------------------
// K1: grayscale + per-image 32-bin histogram (LDS hist -> global atomics)
// ---------------------------------------------------------------------------
__global__ void k_gray_hist(const float* __restrict__ rgb,
                            float* __restrict__ gray,
                            int* __restrict__ hist){
  __shared__ int lh[32];
  int tid = threadIdx.x;
  if (tid < 32) lh[tid] = 0;
  __syncthreads();
  int b = blockIdx.y;
  const float* rp = rgb + (size_t)b*3*HW;
  const float* gp = rp + HW;
  const float* bp = rp + 2*HW;
  float* go = gray + (size_t)b*HW;
#pragma unroll
  for (int i = 0; i < 4; ++i){
    int idx = blockIdx.x*1024 + i*256 + tid;
    __builtin_prefetch(rp + idx + 4096, 0, 1);     // global_prefetch_b8
    float v = 0.5f*(0.299f*rp[idx] + 0.587f*gp[idx] + 0.114f*bp[idx] + 1.0f);
    v = fminf(fmaxf(v, 0.0f), 1.0f);
    go[idx] = v;
    int bin = (int)(v*32.0f);
    bin = bin < 0 ? 0 : (bin > 31 ? 31 : bin);
    atomicAdd(&lh[bin], 1);
  }
  __syncthreads();
  if (tid < 32) atomicAdd(&hist[b*32 + tid], lh[tid]);
}

// ---------------------------------------------------------------------------
// K2: Otsu threshold per image (tiny; serial over 32 bins)
// ---------------------------------------------------------------------------
__global__ void k_otsu(const int* __restrict__ hist, float* __restrict__ thr){
  int b = blockIdx.x;
  if (threadIdx.x == 0){
    float omA[32], muA[32];
    float om = 0.f, mu = 0.f;
    for (int k = 0; k < 32; ++k){
      float p = (float)hist[b*32 + k] / ((float)HW + 1e-6f);
      om += p;
      mu += p * ((float)k / 31.0f);
      omA[k] = om; muA[k] = mu;
    }
    float mu_t = mu;
    float best = -1e30f; int kb = 0;
    for (int k = 0; k < 32; ++k){
      float w = omA[k];
      float d = mu_t*w - muA[k];
      float s = d*d / (w*(1.0f - w) + 1e-8f);
      if (s > best){ best = s; kb = k; }
    }
    thr[b] = (float)kb / 32.0f;   // edges = linspace(0,1,33)
  }
}

// ---------------------------------------------------------------------------
// K3: 4-stage morphology fused in LDS (halo 4). Border = ignore-OOB
// (min pads +inf, max pads -inf) -> per-read coordinate guards.
// ---------------------------------------------------------------------------
__device__ __forceinline__ void morph33(const float* src, float* dst,
                                        int lo, int hi, bool isMin,
                                        int by0, int bx0, int tid){
  int n = hi - lo;
  for (int i = tid; i < n*n; i += 256){
    int ly = lo + i/n, lx = lo + i%n;
    float v = isMin ? 1e30f : -1e30f;
#pragma unroll
    for (int dy = -1; dy <= 1; ++dy)
#pragma unroll
      for (int dx = -1; dx <= 1; ++dx){
        int yy = ly+dy, xx = lx+dx;
        unsigned gy = (unsigned)(by0+yy), gx = (unsigned)(bx0+xx);
        if (gy < HH && gx < WW){
          float s = src[yy*40 + xx];
          v = isMin ? fminf(v, s) : fmaxf(v, s);
        }
      }
    dst[ly*40 + lx] = v;
  }
}

__global__ void k_morph(const float* __restrict__ gray,
                        const float* __restrict__ thr,
                        float* __restrict__ nuc){
  __shared__ float ta[40*40], tb[40*40];
  int tid = threadIdx.x, b = blockIdx.z;
  int bx0 = blockIdx.x*32 - 4, by0 = blockIdx.y*32 - 4;
  float th = thr[b];
  const float* g = gray + (size_t)b*HW;
  for (int i = tid; i < 1600; i += 256){
    int ly = i/40, lx = i%40;
    unsigned gy = (unsigned)(by0+ly), gx = (unsigned)(bx0+lx);
    float v = 0.f;
    if (gy < HH && gx < WW) v = (g[gy*WW+gx] <= th) ? 1.f : 0.f;
    ta[i] = v;
  }
  __syncthreads();
  morph33(ta, tb, 1, 39, true,  by0, bx0, tid); __syncthreads();
  morph33(tb, ta, 2, 38, false, by0, bx0, tid); __syncthreads();
  morph33(ta, tb, 3, 37, false, by0, bx0, tid); __syncthreads();
  morph33(tb, ta, 4, 36, true,  by0, bx0, tid); __syncthreads();
  float* np = nuc + (size_t)b*HW;
  for (int i = tid; i < 1024; i += 256){
    int oy = i/32, ox = i%32;
    int gy = by0+4+oy, gx = bx0+4+ox;
    np[gy*WW+gx] = (ta[(4+oy)*40 + 4+ox] > 0.5f) ? 1.f : 0.f;
  }
}

// ---------------------------------------------------------------------------
// K4: boundary + laplacian + 5x5 local sum -> roughness (z_maps ch 0)
// ---------------------------------------------------------------------------
__global__ void k_rough(const float* __restrict__ nuc, float* __restrict__ zm){
  __shared__ float nt[38*38];
  __shared__ float bnd[36*36];
  __shared__ float hs[36*32];
  int tid = threadIdx.x, b = blockIdx.z;
  int bx0 = blockIdx.x*32 - 3, by0 = blockIdx.y*32 - 3;
  const float* np = nuc + (size_t)b*HW;
  for (int i = tid; i < 38*38; i += 256){
    int ly = i/38, lx = i%38;
    unsigned gy = (unsigned)(by0+ly), gx = (unsigned)(bx0+lx);
    nt[i] = (gy < HH && gx < WW) ? np[gy*WW+gx] : 0.f;   // zero-pad (conv + sums)
  }
  __syncthreads();
  // boundary = nuc - min3(nuc) (min ignores OOB)
  for (int i = tid; i < 36*36; i += 256){
    int ly = 1 + i/36, lx = 1 + i%36;
    unsigned gy = (unsigned)(by0+ly), gx = (unsigned)(bx0+lx);
    float v = 0.f;
    if (gy < HH && gx < WW){
      float er = 1e30f;
#pragma unroll
      for (int dy = -1; dy <= 1; ++dy)
#pragma unroll
        for (int dx = -1; dx <= 1; ++dx){
          unsigned yy = (unsigned)(by0+ly+dy), xx = (unsigned)(bx0+lx+dx);
          if (yy < HH && xx < WW) er = fminf(er, nt[(ly+dy)*38 + lx+dx]);
        }
      v = fmaxf(nt[ly*38+lx] - er, 0.f);
    }
    bnd[i] = v;
  }
  __syncthreads();
  // horizontal 5-sum
  for (int i = tid; i < 36*32; i += 256){
    int r = i/32, c = i%32;
    float s = 0.f;
#pragma unroll
    for (int dx = 0; dx < 5; ++dx) s += bnd[r*36 + c + dx];
    hs[i] = s;
  }
  __syncthreads();
  float* z0 = zm + ((size_t)(b*4 + 0))*HW;
  for (int i = tid; i < 1024; i += 256){
    int oy = i/32, ox = i%32;
    int ly = 3+oy, lx = 3+ox;
    int gy = by0+ly, gx = bx0+lx;
    float den = 1e-6f;
#pragma unroll
    for (int dy = 0; dy < 5; ++dy) den += hs[(oy+dy)*32 + ox];
    float c0 = nt[ly*38+lx];
    float lap = fabsf(nt[(ly-1)*38+lx] + nt[(ly+1)*38+lx]
                    + nt[ly*38+lx-1] + nt[ly*38+lx+1] - 4.f*c0);
    float bd = bnd[(ly-1)*36 + (lx-1)];
    z0[gy*WW+gx] = lap*bd/den;
  }
}

// ---------------------------------------------------------------------------
// K5: cell = maxpool11(nuc); nc = sum9(nuc)/max(sum9(cell)-sum9(nuc),1)
// Vertical 9-sums on the matrix pipe (WMMA). z_maps ch 2.
// ---------------------------------------------------------------------------
__global__ void k_cellnc(const float* __restrict__ nuc, float* __restrict__ zm){
  __shared__ float nt[50*50];
  __shared__ float hm[50*40];
  __shared__ float ct[40*40];
  __shared__ float hsN[40*32];
  __shared__ float hsC[40*32];
  __shared__ float sn[32*32];
  __shared__ float sc[32*32];
  int tid = threadIdx.x, b = blockIdx.z;
  int bx0 = blockIdx.x*32 - 9, by0 = blockIdx.y*32 - 9;
  const float* np = nuc + (size_t)b*HW;
  for (int i = tid; i < 2500; i += 256){
    int ly = i/50, lx = i%50;
    unsigned gy = (unsigned)(by0+ly), gx = (unsigned)(bx0+lx);
    nt[i] = (gy < HH && gx < WW) ? np[gy*WW+gx] : 0.f;  // 0 OK: binary max & zero-pad sums
  }
  __syncthreads();
  // horizontal max11
  for (int i = tid; i < 2000; i += 256){
    int ly = i/40, lx = 5 + i%40;
    float m = 0.f;
#pragma unroll
    for (int dx = 0; dx < 11; ++dx) m = fmaxf(m, nt[ly*50 + lx-5+dx]);
    hm[i] = m;
  }
  __syncthreads();
  // vertical max11 -> cell (zeroed at OOB positions: sums zero-pad cell)
  for (int i = tid; i < 1600; i += 256){
    int r = i/40, c = i%40;   // ly=5+r, lx=5+c
    float m = 0.f;
#pragma unroll
    for (int dy = 0; dy < 11; ++dy) m = fmaxf(m, hm[(r+dy)*40 + c]);
    unsigned gy = (unsigned)(by0+5+r), gx = (unsigned)(bx0+5+c);
    ct[i] = (gy < HH && gx < WW) ? m : 0.f;
  }
  __syncthreads();
  // horizontal 9-sums of nuc and cell
  for (int i = tid; i < 1280; i += 256){
    int r = i/32, c = i%32;   // row ly=5+r, center col lx=9+c
    float s1 = 0.f, s2 = 0.f;
#pragma unroll
    for (int dx = 0; dx < 9; ++dx){
      s1 += nt[(5+r)*50 + 5+c+dx];
      s2 += ct[r*40 + c+dx];
    }
    hsN[i] = s1; hsC[i] = s2;
  }
  __syncthreads();
  // vertical 9-sums: 8 WMMA tile-jobs, one per wave (8 waves, wave32)
  {
    int wave = tid >> 5, lane = tid & 31;
    int f = wave >> 2, sub = wave & 3;
    int sy = (sub >> 1)*16, sx = (sub & 1)*16;
    const float* hs = f ? hsC : hsN;
    float* o = f ? sc : sn;
    box9_vert_wmma(hs + sy*32 + sx, 32, o + sy*32 + sx, 32, lane);
  }
  __syncthreads();
  float* z2 = zm + ((size_t)(b*4 + 2))*HW;
  for (int i = tid; i < 1024; i += 256){
    int oy = i/32, ox = i%32;
    int gy = by0+9+oy, gx = bx0+9+ox;
    float An = sn[i], Ac = fmaxf(sc[i] - An, 1.0f);
    z2[gy*WW+gx] = An/Ac;
  }
}

// ---------------------------------------------------------------------------
// K6: Sobel -> atan2 -> sin/cos; 9x9 sums of sin,cos,gray,gray^2 (WMMA
// vertical); circ_var (ch 1) + ent_like (ch 3). den_orient analytic.
// ---------------------------------------------------------------------------
__global__ void k_texture(const float* __restrict__ gray, float* __restrict__ zm){
  __shared__ float gT[42*42];
  __shared__ float sA[40*40];
  __shared__ float cA[40*40];
  __shared__ float h0[40*32], h1[40*32], h2[40*32], h3[40*32];
  __shared__ float o0[32*32], o1[32*32], o2[32*32], o3[32*32];
  int tid = threadIdx.x, b = blockIdx.z;
  int bx0 = blockIdx.x*32 - 5, by0 = blockIdx.y*32 - 5;
  const float* gp = gray + (size_t)b*HW;
  for (int i = tid; i < 1764; i += 256){
    int ly = i/42, lx = i%42;
    unsigned gy = (unsigned)(by0+ly), gx = (unsigned)(bx0+lx);
    gT[i] = (gy < HH && gx < WW) ? gp[gy*WW+gx] : 0.f;   // conv zero-pads
  }
  __syncthreads();
  // sin/cos(atan2(sobel)) -- zeroed at OOB positions (sums zero-pad)
  for (int i = tid; i < 1600; i += 256){
    int ly = 1 + i/40, lx = 1 + i%40;
    unsigned gy = (unsigned)(by0+ly), gx = (unsigned)(bx0+lx);
    float sv = 0.f, cv = 0.f;
    if (gy < HH && gx < WW){
      const float* p = gT + (ly-1)*42 + lx;
      float tl = p[-1],   tc = p[0],  tr = p[1];
      float ml = p[41],               mr = p[43];
      float bl = p[83],   bc = p[84], br = p[85];
      float gxv = (tl + 2.f*ml + bl) - (tr + 2.f*mr + br);
      float gyv = (tl + 2.f*tc + tr) - (bl + 2.f*bc + br);
      float ang = atan2f(gyv, gxv);
      sincosf(ang, &sv, &cv);
    }
    sA[i] = sv; cA[i] = cv;
  }
  __syncthreads();
  // horizontal 9-sums of sin, cos, g, g^2
  for (int i = tid; i < 1280; i += 256){
    int r = i/32, c = i%32;   // row ly=1+r, center col lx=5+c
    float ss = 0.f, scs = 0.f, sg = 0.f, sg2 = 0.f;
#pragma unroll
    for (int dx = 0; dx < 9; ++dx){
      ss  += sA[r*40 + c+dx];
      scs += cA[r*40 + c+dx];
      float g = gT[(r+1)*42 + (1+c+dx)];
      sg += g; sg2 += g*g;
    }
    h0[i] = ss; h1[i] = scs; h2[i] = sg; h3[i] = sg2;
  }
  __syncthreads();
  // vertical 9-sums: 16 WMMA tile-jobs over 8 waves
  {
    int wave = tid >> 5, lane = tid & 31;
    for (int j = wave; j < 16; j += 8){
      int f = j >> 2, sub = j & 3;
      int sy = (sub >> 1)*16, sx = (sub & 1)*16;
      const float* hs = (f==0)?h0:(f==1)?h1:(f==2)?h2:h3;
      float* o = (f==0)?o0:(f==1)?o1:(f==2)?o2:o3;
      box9_vert_wmma(hs + sy*32 + sx, 32, o + sy*32 + sx, 32, lane);
    }
  }
  __syncthreads();
  float* z1 = zm + ((size_t)(b*4 + 1))*HW;
  float* z3 = zm + ((size_t)(b*4 + 3))*HW;
  for (int i = tid; i < 1024; i += 256){
    int oy = i/32, ox = i%32;
    int gy = by0+5+oy, gx = bx0+5+ox;
    int cy = min(gy+4, HH-1) - max(gy-4, 0) + 1;
    int cx = min(gx+4, WW-1) - max(gx-4, 0) + 1;
    float den = (float)(cy*cx) + 1e-6f;
    float ms = o0[i]/den, mc = o1[i]/den;
    float circ = 1.f - sqrtf(ms*ms + mc*mc + 1e-6f);
    float g1 = o2[i]/den, g2 = o3[i]/den;
    float ent = log1pf(fmaxf(g2 - g1*g1, 0.f));
    z1[gy*WW+gx] = circ;
    z3[gy*WW+gx] = ent;
  }
}

// ---------------------------------------------------------------------------
// K7: stabilize + calibrate + alpha-weighted energy -> E_t
// ---------------------------------------------------------------------------
__global__ void k_energy(const float* __restrict__ zm,
                         const float* __restrict__ cal_a,
                         const float* __restrict__ cal_b,
                         const float* __restrict__ al,
                         float* __restrict__ E){
  int b = blockIdx.y;
  int p = blockIdx.x*256 + threadIdx.x;
  float mx = fmaxf(fmaxf(al[0], al[1]), fmaxf(al[2], al[3]));
  float w[4], sa[4], sb[4], se = 0.f;
#pragma unroll
  for (int c = 0; c < 4; ++c){
    w[c] = expf(al[c] - mx); se += w[c];
    sa[c] = softplusf(cal_a[c]); sb[c] = cal_b[c];
  }
  float inv = 1.f/se;
  size_t base = (size_t)b*4*HW;
  float z0 = log1pf(fmaxf(zm[base + 0*(size_t)HW + p], 0.f));
  float z1 = fminf(fmaxf(zm[base + 1*(size_t)HW + p], 0.f), 1.f);
  float z2 = log1pf(fmaxf(zm[base + 2*(size_t)HW + p], 0.f));
  float z3 = fminf(fmaxf(zm[base + 3*(size_t)HW + p], 0.f), 3.4657359028f);
  float e = w[0]*inv*(sa[0]*z0 + sb[0]) + w[1]*inv*(sa[1]*z1 + sb[1])
          + w[2]*inv*(sa[2]*z2 + sb[2]) + w[3]*inv*(sa[3]*z3 + sb[3]);
  E[(size_t)b*HW + p] = e;
}

// ---------------------------------------------------------------------------
// K8: per-image spatial softmax of -E/tau (E stays L2-resident: 1MB/image)
// ---------------------------------------------------------------------------
__global__ void k_softmax(const float* __restrict__ E,
                          const float* __restrict__ taup,
                          float* __restrict__ A){
  __shared__ float red[1024];
  int tid = threadIdx.x, b = blockIdx.x;
  const float* e = E + (size_t)b*HW;
  float* a = A + (size_t)b*HW;
  float tau = 0.2f + softplusf(taup[0]);
  float nf = -1.0f/tau;
  float m = -1e30f;
  for (int i = tid; i < HW; i += 1024) m = fmaxf(m, e[i]*nf);
  red[tid] = m; __syncthreads();
  for (int off = 512; off > 0; off >>= 1){
    if (tid < off) red[tid] = fmaxf(red[tid], red[tid+off]);
    __syncthreads();
  }
  m = red[0]; __syncthreads();
  float s = 0.f;
  for (int i = tid; i < HW; i += 1024) s += expf(e[i]*nf - m);
  red[tid] = s; __syncthreads();
  for (int off = 512; off > 0; off >>= 1){
    if (tid < off) red[tid] += red[tid+off];
    __syncthreads();
  }
  float rZ = 1.0f/red[0];
  for (int i = tid; i < HW; i += 1024) a[i] = expf(e[i]*nf - m)*rZ;
}

// ---------------------------------------------------------------------------
extern "C" void kernel_launch(void* const* d_in, const int* in_sizes, int n_in,
                              void* d_out, int out_size, void* d_ws, size_t ws_size,
                              hipStream_t stream){
  (void)in_sizes; (void)n_in; (void)out_size; (void)ws_size;
  const float* rgb    = (const float*)d_in[0];
  const float* cal_a  = (const float*)d_in[1];
  const float* cal_b  = (const float*)d_in[2];
  const float* alpha  = (const float*)d_in[3];
  const float* tau_p  = (const float*)d_in[4];

  float* out     = (float*)d_out;
  float* A_out   = out;                          // (16,1,512,512)
  float* zm_out  = out + (size_t)NB*HW;          // (16,4,512,512)
  float* nuc_out = out + (size_t)NB*HW*5;        // (16,1,512,512)
  float* E_out   = out + (size_t)NB*HW*6;        // (16,1,512,512)

  char* ws = (char*)d_ws;
  float* gray = (float*)ws;                                   // 16 MB
  int*   hist = (int*)(ws + (size_t)NB*HW*sizeof(float));     // 2 KB
  float* thr  = (float*)(ws + (size_t)NB*HW*sizeof(float) + NB*32*sizeof(int));

  hipMemsetAsync(hist, 0, NB*32*sizeof(int), stream);

  k_gray_hist<<<dim3(256, NB), 256, 0, stream>>>(rgb, gray, hist);
  k_otsu<<<NB, 32, 0, stream>>>(hist, thr);

  dim3 tg(WW/32, HH/32, NB);
  k_morph  <<<tg, 256, 0, stream>>>(gray, thr, nuc_out);
  k_rough  <<<tg, 256, 0, stream>>>(nuc_out, zm_out);
  k_cellnc <<<tg, 256, 0, stream>>>(nuc_out, zm_out);
  k_texture<<<tg, 256, 0, stream>>>(gray, zm_out);

  k_energy<<<dim3(HW/256, NB), 256, 0, stream>>>(zm_out, cal_a, cal_b, alpha, E_out);
  k_softmax<<<NB, 1024, 0, stream>>>(E_out, tau_p, A_out);
}